// QuantActMSE_44049184588139
// MI455X (gfx1250) — compile-verified
//
#include <hip/hip_runtime.h>
#include <hip/hip_bf16.h>
#include <stdint.h>

// ---------------------------------------------------------------- constants
#define N_TOTAL (32 * 512 * 32 * 32)   // 16,777,216 elements
#define N4      (N_TOTAL / 4)          // 4,194,304 float4
#define NCAND   80
#define TILE    4096                    // floats per LDS tile (16 KB)
#define TILE4   (TILE / 4)              // 1024 float4 per tile
#define NTILES  (N_TOTAL / TILE)        // 4096 tiles (exact)
#define NBLK1   1024                    // min/max partial blocks
#define NBLK3   512                     // scoring blocks
#define NWAVES  8                       // 256 threads = 8 wave32
#define NC      (NCAND / NWAVES)        // 10 candidates per wave

// workspace layout (float offsets)
#define WS_PMIN   0
#define WS_PMAX   (WS_PMIN + NBLK1)                 // 1024
#define WS_MINMAX (WS_PMAX + NBLK1)                 // 2048 (2 floats)
#define WS_PARAMS (WS_MINMAX + 4)                   // 2052 (80 x float4, 16B aligned)
#define WS_SCORES (WS_PARAMS + NCAND * 4)           // 2372 (NBLK3*80 floats)
#define WS_FINAL  (WS_SCORES + NBLK3 * NCAND)       // 43332 (float4, 16B aligned)

typedef __attribute__((ext_vector_type(2))) float v2f;
typedef __attribute__((ext_vector_type(8))) float v8f;

// ------------------------------------------------------- CDNA5 async macros
#define ASYNC_WAIT(n) asm volatile("s_wait_asynccnt %0" :: "i"(n) : "memory")

__device__ __forceinline__ void async_tile_load(uint32_t lds_base_byte,
                                                const float* gsrc, int tid) {
  // 256 threads x 4 x b128 = 4096 floats, tracked on ASYNCcnt
  uint64_t g = (uint64_t)(uintptr_t)gsrc + (uint64_t)tid * 16u;
  uint32_t l = lds_base_byte + (uint32_t)tid * 16u;
#pragma unroll
  for (int k = 0; k < TILE4 / 256; ++k) {   // 4 async b128 per thread
    asm volatile("global_load_async_to_lds_b128 %0, %1, off"
                 :: "v"(l), "v"(g) : "memory");
    g += 256u * 16u;
    l += 256u * 16u;
  }
}

// ------------------------------------------------ wave32 reduction helpers
__device__ __forceinline__ float wave_min32(float v) {
#pragma unroll
  for (int m = 16; m >= 1; m >>= 1) v = fminf(v, __shfl_xor(v, m, 32));
  return v;
}
__device__ __forceinline__ float wave_max32(float v) {
#pragma unroll
  for (int m = 16; m >= 1; m >>= 1) v = fmaxf(v, __shfl_xor(v, m, 32));
  return v;
}

// Full 32-lane sum. WMMA trick: A = (v, 0) -> A[m][0]=v(m), A[m][2]=v(m+16);
// B = all-ones (layout independent) => D[m][n] = v(m)+v(m+16) = S(m).
// D vgpr r: lanes0-15 = S(r), lanes16-31 = S(r+8). Sum 8 vgprs + xor16 = total.
__device__ __forceinline__ float wave_sum32(float v) {
#if __has_builtin(__builtin_amdgcn_wmma_f32_16x16x4_f32)
  v2f a; a[0] = v; a[1] = 0.0f;
  v2f b; b[0] = 1.0f; b[1] = 1.0f;
  v8f c = {};
  v8f d = __builtin_amdgcn_wmma_f32_16x16x4_f32(false, a, false, b,
                                                (short)0, c, false, false);
  float s = ((d[0] + d[1]) + (d[2] + d[3])) + ((d[4] + d[5]) + (d[6] + d[7]));
  s += __shfl_xor(s, 16, 32);
  return s;
#else
#pragma unroll
  for (int m = 16; m >= 1; m >>= 1) v += __shfl_xor(v, m, 32);
  return v;
#endif
}

// ------------------------------------------------- kernel 1: min/max partials
__global__ __launch_bounds__(256) void k_minmax(const float4* __restrict__ x,
                                                float* __restrict__ ws) {
  const int tid = threadIdx.x;
  float mn = INFINITY, mx = -INFINITY;
  for (int i = blockIdx.x * 256 + tid; i < N4; i += NBLK1 * 256) {
    float4 v = x[i];
    mn = fminf(mn, fminf(fminf(v.x, v.y), fminf(v.z, v.w)));
    mx = fmaxf(mx, fmaxf(fmaxf(v.x, v.y), fmaxf(v.z, v.w)));
  }
  mn = wave_min32(mn);
  mx = wave_max32(mx);
  __shared__ float smn[NWAVES], smx[NWAVES];
  if ((tid & 31) == 0) { smn[tid >> 5] = mn; smx[tid >> 5] = mx; }
  __syncthreads();
  if (tid == 0) {
    float bm = smn[0], bM = smx[0];
#pragma unroll
    for (int w = 1; w < NWAVES; ++w) { bm = fminf(bm, smn[w]); bM = fmaxf(bM, smx[w]); }
    ws[WS_PMIN + blockIdx.x] = bm;
    ws[WS_PMAX + blockIdx.x] = bM;
  }
}

// --------------------------- kernel 2: finalize min/max + 80 candidate params
__global__ __launch_bounds__(256) void k_setup(float* __restrict__ ws) {
  const int tid = threadIdx.x;
  float mn = INFINITY, mx = -INFINITY;
#pragma unroll
  for (int k = 0; k < NBLK1 / 256; ++k) {
    mn = fminf(mn, ws[WS_PMIN + tid + k * 256]);
    mx = fmaxf(mx, ws[WS_PMAX + tid + k * 256]);
  }
  mn = wave_min32(mn);
  mx = wave_max32(mx);
  __shared__ float smn[NWAVES], smx[NWAVES];
  __shared__ float gmn, gmx;
  if ((tid & 31) == 0) { smn[tid >> 5] = mn; smx[tid >> 5] = mx; }
  __syncthreads();
  if (tid == 0) {
    float bm = smn[0], bM = smx[0];
#pragma unroll
    for (int w = 1; w < NWAVES; ++w) { bm = fminf(bm, smn[w]); bM = fmaxf(bM, smx[w]); }
    gmn = bm; gmx = bM;
    ws[WS_MINMAX + 0] = bm;
    ws[WS_MINMAX + 1] = bM;
  }
  __syncthreads();
  if (tid < NCAND) {
    float f    = 1.0f - (float)tid * 0.01f;
    float cmn  = gmn * f;
    float cmx  = gmx * f;
    float delta = fmaxf(cmx - cmn, 1e-8f) / 255.0f;
    float zp    = rintf(-cmn / delta);
    float4* p = (float4*)(ws + WS_PARAMS);
    p[tid] = make_float4(1.0f / delta, delta, -zp, 255.0f - zp);
  }
}

// ------- kernel 3: fused 80-candidate Lp scoring (async-LDS double buffered)
__global__ __launch_bounds__(256) void k_score(const float* __restrict__ x,
                                               float* __restrict__ ws) {
  __shared__ float lds[2][TILE];
  const int tid  = threadIdx.x;
  const int wave = tid >> 5;
  const int lane = tid & 31;

  // this wave's 10 candidate param sets, hoisted to registers
  const float4* params = (const float4*)(ws + WS_PARAMS);
  float invd[NC], dlt[NC], lo[NC], hi[NC];
#pragma unroll
  for (int c = 0; c < NC; ++c) {
    float4 p = params[wave * NC + c];
    invd[c] = p.x; dlt[c] = p.y; lo[c] = p.z; hi[c] = p.w;
  }
  float acc[NC];
#pragma unroll
  for (int c = 0; c < NC; ++c) acc[c] = 0.0f;

  const uint32_t lbase0 = (uint32_t)(uintptr_t)(&lds[0][0]);
  const uint32_t lbase1 = (uint32_t)(uintptr_t)(&lds[1][0]);

  int tile = blockIdx.x;
  async_tile_load(lbase0, x + (size_t)tile * TILE, tid);
  int buf = 0;
  for (; tile < NTILES; tile += NBLK3) {
    const int nxt = tile + NBLK3;
    if (nxt < NTILES) {
      async_tile_load(buf ? lbase0 : lbase1, x + (size_t)nxt * TILE, tid);
      ASYNC_WAIT(4);   // our 4 current-tile b128s done; next tile's 4 in flight
    } else {
      ASYNC_WAIT(0);
    }
    __syncthreads();   // everyone waited own asynccnt -> tile fully in LDS

    const float4* lb = (const float4*)(&lds[buf][0]);
    for (int j = lane; j < TILE4; j += 32) {
      float4 t4 = lb[j];
      float te[4] = {t4.x, t4.y, t4.z, t4.w};
#pragma unroll
      for (int e = 0; e < 4; ++e) {
        const float t = te[e];
#pragma unroll
        for (int c = 0; c < NC; ++c) {
          float q = rintf(t * invd[c]);
          q = fminf(fmaxf(q, lo[c]), hi[c]);
          float err = __builtin_fmaf(q, dlt[c], -t);   // dequant - t
          float s2  = err * err;
          // |err|^2.4 = exp2(1.2 * log2(err^2)); err==0 -> exp2(-inf)=0
          acc[c] += __builtin_amdgcn_exp2f(1.2f * __builtin_amdgcn_logf(s2));
        }
      }
    }
    __syncthreads();   // done reading buf before it gets overwritten
    buf ^= 1;
  }

  // per-candidate wave-wide sum via v_wmma_f32_16x16x4_f32 reduction
#pragma unroll
  for (int c = 0; c < NC; ++c) {
    float tot = wave_sum32(acc[c]);
    if (lane == 0)
      ws[WS_SCORES + (size_t)blockIdx.x * NCAND + wave * NC + c] = tot;
  }
}

// --------------- kernel 4: reduce scores, argmin (tie->first), final params
__global__ __launch_bounds__(128) void k_argmin(float* __restrict__ ws,
                                                const float* __restrict__ xminbuf,
                                                const float* __restrict__ xmaxbuf) {
  __shared__ float ssc[NCAND];
  const int t = threadIdx.x;
  if (t < NCAND) {
    float s = 0.0f;
    for (int b = 0; b < NBLK3; ++b) s += ws[WS_SCORES + (size_t)b * NCAND + t];
    ssc[t] = s;
  }
  __syncthreads();
  if (t == 0) {
    int best = 0; float bs = ssc[0];
    for (int i = 1; i < NCAND; ++i)
      if (ssc[i] < bs) { bs = ssc[i]; best = i; }   // strict '<' => first tie wins
    const float factor   = 1.0f - (float)best * 0.01f;
    const float save_min = ws[WS_MINMAX + 0] * factor;
    const float save_max = ws[WS_MINMAX + 1] * factor;
    const float new_min  = xminbuf[0] * 0.9f + save_min * 0.1f;
    const float new_max  = xmaxbuf[0] * 0.9f + save_max * 0.1f;
    const float delta    = fmaxf(new_max - new_min, 1e-8f) / 255.0f;
    const float zp       = rintf(-new_min / delta);
    float4* fp = (float4*)(ws + WS_FINAL);
    fp[0] = make_float4(1.0f / delta, delta, -zp, 255.0f - zp);
  }
}

// ------------------------------- kernel 5: final quant-dequant (streaming)
__global__ __launch_bounds__(256) void k_quant(const float4* __restrict__ x,
                                               float4* __restrict__ out,
                                               const float* __restrict__ ws) {
  const float4 p = *(const float4*)(ws + WS_FINAL);  // {invd, delta, lo, hi}
  const int i = blockIdx.x * 256 + threadIdx.x;
  if (i < N4) {
    float4 v = x[i];
    float4 r;
    r.x = fminf(fmaxf(rintf(v.x * p.x), p.z), p.w) * p.y;
    r.y = fminf(fmaxf(rintf(v.y * p.x), p.z), p.w) * p.y;
    r.z = fminf(fmaxf(rintf(v.z * p.x), p.z), p.w) * p.y;
    r.w = fminf(fmaxf(rintf(v.w * p.x), p.z), p.w) * p.y;
    out[i] = r;
  }
}

// --------------------------------------------------------------- launcher
extern "C" void kernel_launch(void* const* d_in, const int* in_sizes, int n_in,
                              void* d_out, int out_size, void* d_ws, size_t ws_size,
                              hipStream_t stream) {
  (void)in_sizes; (void)n_in; (void)out_size; (void)ws_size;
  const float*  x        = (const float*)d_in[0];
  const float*  xmin_buf = (const float*)d_in[1];
  const float*  xmax_buf = (const float*)d_in[2];
  float*        out      = (float*)d_out;
  float*        ws       = (float*)d_ws;

  k_minmax<<<NBLK1, 256, 0, stream>>>((const float4*)x, ws);
  k_setup <<<1,     256, 0, stream>>>(ws);
  k_score <<<NBLK3, 256, 0, stream>>>(x, ws);
  k_argmin<<<1,     128, 0, stream>>>(ws, xmin_buf, xmax_buf);
  k_quant <<<N4 / 256, 256, 0, stream>>>((const float4*)x, (float4*)out, ws);
}